// Net_57887569215769
// MI455X (gfx1250) — compile-verified
//
#include <hip/hip_runtime.h>
#include <hip/hip_bf16.h>

#define N_NODES   100000
#define N_EDGES   3200000
#define D_FEAT    512
#define D_HID     16
#define N_CLASSES 7

typedef float v2f __attribute__((ext_vector_type(2)));
typedef float v8f __attribute__((ext_vector_type(8)));

// ---------------------------------------------------------------------------
// Init: zero agg1, seed out with b2 (out = segment_sum(...) + b2)
// ---------------------------------------------------------------------------
__global__ void init_kernel(float* __restrict__ agg1,
                            float* __restrict__ out,
                            const float* __restrict__ b2) {
  int i = blockIdx.x * blockDim.x + threadIdx.x;
  if (i < N_NODES * D_HID) agg1[i] = 0.0f;
  if (i < N_NODES * N_CLASSES) out[i] = b2[i % N_CLASSES];
}

// ---------------------------------------------------------------------------
// GEMM1: t1[100000,16] = features[100000,512] @ W1[512,16]
// One wave per 16-node tile, V_WMMA_F32_16X16X4_F32, K-loop of 128 steps.
// W1 staged in LDS (32 KB). Bandwidth-bound on the 204.8 MB feature stream.
// ---------------------------------------------------------------------------
__global__ void gemm1_wmma(const float* __restrict__ feat,
                           const float* __restrict__ W1,
                           float* __restrict__ t1) {
  __shared__ float ldsW[D_FEAT * D_HID];  // 8192 f32 = 32 KB

  // cooperative W1 load (256 threads x 8 float4 each)
  {
    const float4* s4 = (const float4*)W1;
    float4* d4 = (float4*)ldsW;
    for (int i = threadIdx.x; i < (D_FEAT * D_HID) / 4; i += blockDim.x)
      d4[i] = s4[i];
  }
  __syncthreads();

  const int lane = threadIdx.x & 31;
  const int wave = threadIdx.x >> 5;
  const int tile = blockIdx.x * (blockDim.x >> 5) + wave;
  if (tile >= N_NODES / 16) return;   // wave-uniform guard: EXEC stays all-1s
  const int node0 = tile * 16;

  // A layout (16x4 f32, MxK): lanes 0-15 hold M=lane, K={k0,k0+1};
  //                           lanes 16-31 hold M=lane-16, K={k0+2,k0+3}
  // B layout (4x16 f32, KxN): lanes 0-15 hold N=lane, K={k0,k0+1};
  //                           lanes 16-31 hold N=lane-16, K={k0+2,k0+3}
  const int mn   = lane & 15;               // M for A, N for B
  const int koff = (lane >> 4) << 1;        // 0 or 2
  const float* arow = feat + (size_t)(node0 + mn) * D_FEAT + koff;
  const float* bcol = ldsW + koff * D_HID + mn;

  v8f acc = {};
  #pragma unroll 4
  for (int kk = 0; kk < D_FEAT; kk += 4) {
    v2f a = *(const v2f*)(arow + kk);
    v2f b;
    b.x = bcol[kk * D_HID];
    b.y = bcol[kk * D_HID + D_HID];
    // (neg_a, A, neg_b, B, c_mod, C, reuse_a, reuse_b)
    acc = __builtin_amdgcn_wmma_f32_16x16x4_f32(
        false, a, false, b, (short)0, acc, false, false);
  }

  // C/D layout: VGPR r -> lanes 0-15: M=r, N=lane; lanes 16-31: M=r+8, N=lane-16
  const int rbase = node0 + ((lane >> 4) << 3);
  #pragma unroll
  for (int r = 0; r < 8; ++r)
    t1[(size_t)(rbase + r) * D_HID + mn] = acc[r];
}

// ---------------------------------------------------------------------------
// Scatter 1: agg1[dst] += t1[src]  (16 floats / edge, L2-resident atomics)
// ---------------------------------------------------------------------------
__global__ void scatter1(const float* __restrict__ t1,
                         const int* __restrict__ src,
                         const int* __restrict__ dst,
                         float* __restrict__ agg1) {
  int e = blockIdx.x * blockDim.x + threadIdx.x;
  if (e >= N_EDGES) return;
  const int s = src[e];
  const int d = dst[e];
  const float4* hv = (const float4*)(t1 + (size_t)s * D_HID);
  float* av = agg1 + (size_t)d * D_HID;
  #pragma unroll
  for (int q = 0; q < 4; ++q) {
    float4 v = hv[q];
    atomicAdd(av + q * 4 + 0, v.x);
    atomicAdd(av + q * 4 + 1, v.y);
    atomicAdd(av + q * 4 + 2, v.z);
    atomicAdd(av + q * 4 + 3, v.w);
  }
}

// ---------------------------------------------------------------------------
// Layer 2 transform: h = relu(agg1 + b1); t2 = h @ W2[16,7]   (tiny)
// ---------------------------------------------------------------------------
__global__ void layer2_transform(const float* __restrict__ agg1,
                                 const float* __restrict__ b1,
                                 const float* __restrict__ W2,
                                 float* __restrict__ t2) {
  int node = blockIdx.x * blockDim.x + threadIdx.x;
  if (node >= N_NODES) return;
  float h[D_HID];
  const float4* a4 = (const float4*)(agg1 + (size_t)node * D_HID);
  #pragma unroll
  for (int q = 0; q < 4; ++q) {
    float4 v = a4[q];
    h[q * 4 + 0] = fmaxf(v.x + b1[q * 4 + 0], 0.0f);
    h[q * 4 + 1] = fmaxf(v.y + b1[q * 4 + 1], 0.0f);
    h[q * 4 + 2] = fmaxf(v.z + b1[q * 4 + 2], 0.0f);
    h[q * 4 + 3] = fmaxf(v.w + b1[q * 4 + 3], 0.0f);
  }
  #pragma unroll
  for (int c = 0; c < N_CLASSES; ++c) {
    float o = 0.0f;
    #pragma unroll
    for (int j = 0; j < D_HID; ++j)
      o = fmaf(h[j], W2[j * N_CLASSES + c], o);
    t2[(size_t)node * N_CLASSES + c] = o;
  }
}

// ---------------------------------------------------------------------------
// Scatter 2: out[dst] += t2[src]  (7 floats / edge; out pre-seeded with b2)
// ---------------------------------------------------------------------------
__global__ void scatter2(const float* __restrict__ t2,
                         const int* __restrict__ src,
                         const int* __restrict__ dst,
                         float* __restrict__ out) {
  int e = blockIdx.x * blockDim.x + threadIdx.x;
  if (e >= N_EDGES) return;
  const int s = src[e];
  const int d = dst[e];
  const float* ts = t2 + (size_t)s * N_CLASSES;
  float* od = out + (size_t)d * N_CLASSES;
  #pragma unroll
  for (int c = 0; c < N_CLASSES; ++c)
    atomicAdd(od + c, ts[c]);
}

// ---------------------------------------------------------------------------
extern "C" void kernel_launch(void* const* d_in, const int* in_sizes, int n_in,
                              void* d_out, int out_size, void* d_ws, size_t ws_size,
                              hipStream_t stream) {
  const float* feat = (const float*)d_in[0];   // [100000,512]
  const float* W1   = (const float*)d_in[1];   // [512,16]
  const float* b1   = (const float*)d_in[2];   // [16]
  const float* W2   = (const float*)d_in[3];   // [16,7]
  const float* b2   = (const float*)d_in[4];   // [7]
  const int*   src  = (const int*)d_in[5];     // [3.2M]
  const int*   dst  = (const int*)d_in[6];     // [3.2M]
  float* out = (float*)d_out;                  // [100000,7]

  float* ws   = (float*)d_ws;
  float* t1   = ws;                                      // 1,600,000 f32
  float* agg1 = ws + (size_t)N_NODES * D_HID;            // 1,600,000 f32
  float* t2   = ws + 2 * (size_t)N_NODES * D_HID;        //   700,000 f32

  const int threads = 256;
  // init covers max(N_NODES*D_HID, N_NODES*N_CLASSES) = 1.6M elements
  init_kernel<<<(N_NODES * D_HID + threads - 1) / threads, threads, 0, stream>>>(
      agg1, out, b2);

  const int tiles = N_NODES / 16;              // 6250 wave-tiles
  const int waves_per_block = threads / 32;    // 8
  gemm1_wmma<<<(tiles + waves_per_block - 1) / waves_per_block, threads, 0, stream>>>(
      feat, W1, t1);

  scatter1<<<(N_EDGES + threads - 1) / threads, threads, 0, stream>>>(
      t1, src, dst, agg1);

  layer2_transform<<<(N_NODES + threads - 1) / threads, threads, 0, stream>>>(
      agg1, b1, W2, t2);

  scatter2<<<(N_EDGES + threads - 1) / threads, threads, 0, stream>>>(
      t2, src, dst, out);
}